// ModShiftModel_55697135895164
// MI455X (gfx1250) — compile-verified
//
#include <hip/hip_runtime.h>

// ---------------------------------------------------------------------------
// Pairwise clipped-linear loss on [N=8192, C=32] fp32 points (MI455X, gfx1250)
//
// loss = 0.25 * sum_{i,j} f(d_ij)
//   f(d) = (2t-1)(1-t), t = min(d/2, 1)
//        = 1.5*d - 0.5*d^2 - 1   for d^2 <= 4, else 0   (continuous at d=2)
//
// d^2 via Gram matrix: d^2 = sq_i + sq_j - 2 * <x_i, x_j>
// Gram computed with V_WMMA_F32_16X16X4_F32 (fp32 matrix path, K=32 -> 8 WMMA
// per 16x16 tile). Symmetry exploited: only j>=i tiles computed; off-diagonal
// tiles counted twice. Block b handles i-rows {b, 511-b} -> 513 tiles/block
// (perfect balance). Raw v_sqrt_f32 (no IEEE fixup chain) — inputs are far
// from the denormal range where the fixup matters.
// ---------------------------------------------------------------------------

typedef float v2f __attribute__((ext_vector_type(2)));
typedef float v8f __attribute__((ext_vector_type(8)));

#define N_PTS 8192
#define N_CH  32
#define TILES (N_PTS / 16)          // 512 tiles per dimension
#define WAVES_PER_BLOCK 8           // 256 threads, wave32
#define BLOCKS (TILES / 2)          // 256 blocks: block b owns i-rows {b, 511-b}

// Prologue: per-point squared norms into d_ws, and zero the scalar output.
__global__ __launch_bounds__(256) void sqnorm_kernel(const float* __restrict__ mv,
                                                     float* __restrict__ sq,
                                                     float* __restrict__ out) {
    int n = blockIdx.x * blockDim.x + threadIdx.x;
    if (n == 0) out[0] = 0.0f;
    if (n < N_PTS) {
        float s = 0.0f;
#pragma unroll
        for (int c = 0; c < N_CH; ++c) {
            float v = mv[c * N_PTS + n];   // [C, N] layout: contiguous in n
            s += v * v;
        }
        sq[n] = s;
    }
}

__global__ __launch_bounds__(256) void pairwise_loss_kernel(const float* __restrict__ mv,
                                                            const float* __restrict__ sq,
                                                            float* __restrict__ out) {
    const int lane = threadIdx.x & 31;
    const int wave = threadIdx.x >> 5;        // 0..7
    const int m    = lane & 15;               // row/col within tile
    const int kh   = lane >> 4;               // K half-select (ISA f32 A/B layout)

    float lsum = 0.0f;

#pragma unroll 1
    for (int half = 0; half < 2; ++half) {
        const int it = (half == 0) ? (int)blockIdx.x : (TILES - 1 - (int)blockIdx.x);
        const int i0 = it * 16;

        // A fragment: 16x32 fp32 tile of rows i0..i0+15, 8 K=4 steps.
        // ISA 16x4 f32 A layout: lane L holds M=L%16; VGPR0={K0|K2}, VGPR1={K1|K3}
        // by lane half. mv is [C, N] so address = mv[k*N + i0 + m] (64B coalesced).
        v2f afrag[8];
#pragma unroll
        for (int kk = 0; kk < 8; ++kk) {
            const int k0 = 4 * kk + 2 * kh;
            afrag[kk].x = mv[(k0 + 0) * N_PTS + i0 + m];
            afrag[kk].y = mv[(k0 + 1) * N_PTS + i0 + m];
        }

        // Row squared norms matching C/D layout: acc[r] at lane L is element
        // (M = r + 8*(L>=16), N = L%16).
        float sqi[8];
#pragma unroll
        for (int r = 0; r < 8; ++r) sqi[r] = sq[i0 + r + 8 * kh];

        // Upper-triangular j-range, strided across the 8 waves.
        for (int jt = it + wave; jt < TILES; jt += WAVES_PER_BLOCK) {
            const int j0 = jt * 16;

            v8f acc = {};
#pragma unroll
            for (int kk = 0; kk < 8; ++kk) {
                const int k0 = 4 * kk + 2 * kh;
                v2f b;   // B[k][n] = X[j0+n][k] = mv[k*N + j0 + n]
                b.x = mv[(k0 + 0) * N_PTS + j0 + m];
                b.y = mv[(k0 + 1) * N_PTS + j0 + m];
                // 8 args: (neg_a, A, neg_b, B, c_mod, C, reuse_a, reuse_b)
                acc = __builtin_amdgcn_wmma_f32_16x16x4_f32(
                    false, afrag[kk], false, b, (short)0, acc, false, false);
            }

            const float sqj = sq[j0 + m];
            float ts = 0.0f;
#pragma unroll
            for (int r = 0; r < 8; ++r) {
                float d2 = fmaxf(__builtin_fmaf(-2.0f, acc[r], sqi[r] + sqj), 0.0f);
                float d  = __builtin_amdgcn_sqrtf(d2);          // raw v_sqrt_f32
                // f = 1.5*d - 0.5*d2 - 1  (valid for d2 <= 4, 0 beyond)
                float f  = __builtin_fmaf(1.5f, d, __builtin_fmaf(-0.5f, d2, -1.0f));
                ts += (d2 < 4.0f) ? f : 0.0f;
            }
            // Off-diagonal tiles appear twice in the full sum.
            lsum += (jt == it) ? ts : 2.0f * ts;
        }
    }

    // Wave32 butterfly reduction.
#pragma unroll
    for (int off = 16; off >= 1; off >>= 1)
        lsum += __shfl_xor(lsum, off, 32);

    if (lane == 0)
        atomicAdd(out, 0.25f * lsum);
}

extern "C" void kernel_launch(void* const* d_in, const int* in_sizes, int n_in,
                              void* d_out, int out_size, void* d_ws, size_t ws_size,
                              hipStream_t stream) {
    const float* mv  = (const float*)d_in[0];   // [1, 32, 8192] fp32
    float*       out = (float*)d_out;           // scalar
    float*       sq  = (float*)d_ws;            // 8192 floats of scratch

    sqnorm_kernel<<<N_PTS / 256, 256, 0, stream>>>(mv, sq, out);
    pairwise_loss_kernel<<<BLOCKS, 256, 0, stream>>>(mv, sq, out);
}